// GraphSAGEtargetModel_13606456393728
// MI455X (gfx1250) — compile-verified
//
#include <hip/hip_runtime.h>

typedef float v2f __attribute__((ext_vector_type(2)));
typedef float v8f __attribute__((ext_vector_type(8)));

#define NN  100000   // nodes
#define DIN 128      // input feature dim
#define DH  8        // hidden dim
#define NC  40       // classes

// ---------------------------------------------------------------------------
// Zero the degree counter and the 8-wide accumulator.
__global__ __launch_bounds__(256) void k_init(float* __restrict__ cnt,
                                              float* __restrict__ agg, int n) {
  int i = blockIdx.x * blockDim.x + threadIdx.x;
  if (i < n) cnt[i] = 0.0f;
  if (i < n * DH) agg[i] = 0.0f;
}

// ---------------------------------------------------------------------------
// Degree count: cnt[dst] += 1 per edge.
__global__ __launch_bounds__(256) void k_count(const int* __restrict__ ei,
                                               float* __restrict__ cnt, int E) {
  int e = blockIdx.x * blockDim.x + threadIdx.x;
  if (e < E) unsafeAtomicAdd(&cnt[ei[E + e]], 1.0f);
}

// ---------------------------------------------------------------------------
// Fused layer-1 projection via WMMA:  P[i][0..7]  = (x @ W1_l^T)[i]
//                                     P[i][8..15] = (x @ W1_r^T)[i]
// One wave per 16-node tile. D = A(16x4 f32) x B(4x16 f32) + C, K looped 128/4.
// A layout: lanes 0-15 hold K=k,k+1 of row M=lane; lanes 16-31 hold K=k+2,k+3.
// B layout: lanes 0-15 hold rows K=k,k+1 at col N=lane; lanes 16-31 K=k+2,k+3.
// B[k][n] = Wc[n][k] where Wc = [W1_l ; W1_r] (16 rows of 128).
__global__ __launch_bounds__(256) void k_proj1(const float* __restrict__ x,
                                               const float* __restrict__ W1l,
                                               const float* __restrict__ W1r,
                                               float* __restrict__ P) {
  int wave = (blockIdx.x * blockDim.x + threadIdx.x) >> 5;
  int lane = threadIdx.x & 31;
  int ntiles = (NN + 15) >> 4;
  if (wave >= ntiles) return;           // wave-uniform exit: EXEC stays all-1s

  int hi = lane >> 4;                   // 0: K+0/K+1 half, 1: K+2/K+3 half
  int l  = lane & 15;

  int node = wave * 16 + l;
  if (node >= NN) node = NN - 1;        // clamp loads; stores guarded below
  const float* arow = x + (size_t)node * DIN + hi * 2;
  const float* brow = (l < DH ? W1l + (size_t)l * DIN
                              : W1r + (size_t)(l - DH) * DIN) + hi * 2;

  v8f acc = {};
#pragma unroll 4
  for (int k = 0; k < DIN; k += 4) {
    v2f a = *(const v2f*)(arow + k);
    v2f b = *(const v2f*)(brow + k);
    // 8 args: (neg_a, A, neg_b, B, c_mod, C, reuse_a, reuse_b)
    acc = __builtin_amdgcn_wmma_f32_16x16x4_f32(
        false, a, false, b, (short)0, acc, false, false);
  }

  // C/D layout: VGPR v, lanes 0-15 -> M=v, N=lane; lanes 16-31 -> M=v+8.
  int n = l;
  int mbase = wave * 16 + hi * 8;
#pragma unroll
  for (int v = 0; v < 8; ++v) {
    int m = mbase + v;
    if (m < NN) P[(size_t)m * 16 + n] = acc[v];
  }
}

// ---------------------------------------------------------------------------
// Edge scatter: agg[dst][j] += feat[src][j], feat rows have stride 16,
// payload is the first 8 columns. One thread per (edge, j) for coalescing.
__global__ __launch_bounds__(256) void k_scatter(const int* __restrict__ ei,
                                                 const float* __restrict__ feat,
                                                 float* __restrict__ agg, int E) {
  long long tid = (long long)blockIdx.x * blockDim.x + threadIdx.x;
  if (tid >= (long long)E * DH) return;
  int e = (int)(tid >> 3);
  int j = (int)(tid & 7);
  int s = ei[e];
  int d = ei[E + e];
  unsafeAtomicAdd(&agg[(size_t)d * DH + j], feat[(size_t)s * 16 + j]);
}

// ---------------------------------------------------------------------------
// h = relu(agg/deg + P[:,8:16] + b1), stored into P[:,0:8]; agg re-zeroed
// for the second scatter pass.
__global__ __launch_bounds__(256) void k_hidden(const float* __restrict__ cnt,
                                                float* __restrict__ P,
                                                float* __restrict__ agg,
                                                const float* __restrict__ b1) {
  int tid = blockIdx.x * blockDim.x + threadIdx.x;
  if (tid >= NN * DH) return;
  int i = tid >> 3, j = tid & 7;
  float inv = 1.0f / fmaxf(cnt[i], 1.0f);
  float v = agg[tid] * inv + P[(size_t)i * 16 + DH + j] + b1[j];
  P[(size_t)i * 16 + j] = fmaxf(v, 0.0f);
  agg[tid] = 0.0f;
}

// ---------------------------------------------------------------------------
// out[i][c] = sum_j (agg2[i][j]/deg) W2_l[c][j] + h[i][j] W2_r[c][j] + b2[c]
__global__ __launch_bounds__(256) void k_out(const float* __restrict__ P,
                                             const float* __restrict__ agg,
                                             const float* __restrict__ cnt,
                                             const float* __restrict__ W2l,
                                             const float* __restrict__ W2r,
                                             const float* __restrict__ b2,
                                             float* __restrict__ out) {
  int tid = blockIdx.x * blockDim.x + threadIdx.x;
  if (tid >= NN * NC) return;
  int i = tid / NC, c = tid % NC;
  float inv = 1.0f / fmaxf(cnt[i], 1.0f);
  float s = b2[c];
#pragma unroll
  for (int j = 0; j < DH; ++j) {
    s = fmaf(agg[(size_t)i * DH + j] * inv, W2l[c * DH + j], s);
    s = fmaf(P[(size_t)i * 16 + j],        W2r[c * DH + j], s);
  }
  out[tid] = s;
}

// ---------------------------------------------------------------------------
extern "C" void kernel_launch(void* const* d_in, const int* in_sizes, int n_in,
                              void* d_out, int out_size, void* d_ws, size_t ws_size,
                              hipStream_t stream) {
  const float* x   = (const float*)d_in[0];
  const int*   ei  = (const int*)d_in[1];   // [2,E] int32 per harness contract
  const float* W1l = (const float*)d_in[2];
  const float* W1r = (const float*)d_in[3];
  const float* b1  = (const float*)d_in[4];
  const float* W2l = (const float*)d_in[5];
  const float* W2r = (const float*)d_in[6];
  const float* b2  = (const float*)d_in[7];
  float* out = (float*)d_out;

  const int E = in_sizes[1] / 2;

  // Workspace layout (floats): cnt[NN] | P[NN*16] | agg[NN*8]  (~10 MB)
  float* ws  = (float*)d_ws;
  float* cnt = ws;
  float* P   = ws + NN;
  float* agg = ws + NN + (size_t)NN * 16;

  const int tpb = 256;
  const int ntiles = (NN + 15) / 16;

  k_init<<<(NN * DH + tpb - 1) / tpb, tpb, 0, stream>>>(cnt, agg, NN);
  k_count<<<(E + tpb - 1) / tpb, tpb, 0, stream>>>(ei, cnt, E);
  k_proj1<<<(ntiles + 7) / 8, tpb, 0, stream>>>(x, W1l, W1r, P);

  long long sthreads = (long long)E * DH;
  int sblocks = (int)((sthreads + tpb - 1) / tpb);
  k_scatter<<<sblocks, tpb, 0, stream>>>(ei, P, agg, E);       // agg1 += y1[src]
  k_hidden<<<(NN * DH + tpb - 1) / tpb, tpb, 0, stream>>>(cnt, P, agg, b1);
  k_scatter<<<sblocks, tpb, 0, stream>>>(ei, P, agg, E);       // agg2 += h[src]
  k_out<<<(NN * NC + tpb - 1) / tpb, tpb, 0, stream>>>(P, agg, cnt, W2l, W2r, b2, out);
}